// RecurrentJEPA_Model_13469017440896
// MI455X (gfx1250) — compile-verified
//
#include <hip/hip_runtime.h>
#include <math.h>

typedef float v2f __attribute__((ext_vector_type(2)));
typedef float v8f __attribute__((ext_vector_type(8)));
typedef unsigned int u32x4 __attribute__((ext_vector_type(4)));
typedef int i32x8 __attribute__((ext_vector_type(8)));
typedef int i32x4 __attribute__((ext_vector_type(4)));

#define B_        512
#define REPR_     256
#define HID_      512
#define TSTEPS    16
#define XSTRIDE   260                 // 256 (s) + 2 (u) + 2 zero pad, multiple of 4
#define OUT_STRIDE (17 * REPR_)       // 4352

__device__ __forceinline__ v8f wmma4(v2f a, v2f b, v8f c) {
  return __builtin_amdgcn_wmma_f32_16x16x4_f32(false, a, false, b, (short)0, c, false, false);
}

// ---------------------------------------------------------------- utilities
__global__ __launch_bounds__(256) void zero_kernel(float* __restrict__ p, int n) {
  int i = blockIdx.x * 256 + threadIdx.x;
  if (i < n) p[i] = 0.0f;
}

// Pack gru_w_ih (1536 x 258) into (1536 x 260) with zero pad cols 258..259.
__global__ __launch_bounds__(256) void pack_wih_kernel(const float* __restrict__ wih,
                                                       float* __restrict__ wihp) {
  int idx = blockIdx.x * 256 + threadIdx.x;
  const int total = 3 * HID_ * XSTRIDE;
  if (idx >= total) return;
  int row = idx / XSTRIDE, col = idx - row * XSTRIDE;
  wihp[idx] = (col < REPR_ + 2) ? wih[row * (REPR_ + 2) + col] : 0.0f;
}

// ---------------------------------------------------------------- encoder
__global__ __launch_bounds__(256) void conv_stats_kernel(const float* __restrict__ x,
                                                         const float* __restrict__ cw,
                                                         float* __restrict__ partial) {
  __shared__ float sw[288];
  __shared__ float buf[32 * 257];
  const int tid = threadIdx.x;
  const int n   = blockIdx.x;
  for (int i = tid; i < 288; i += 256) sw[i] = cw[i];
  __syncthreads();
  const float* xn = x + (size_t)n * 2 * 4096;
  float sum[16], sq[16];
#pragma unroll
  for (int c = 0; c < 16; ++c) { sum[c] = 0.f; sq[c] = 0.f; }
  for (int p = tid; p < 4096; p += 256) {
    int h = p >> 6, w = p & 63;
    float xin[2][3][3];
#pragma unroll
    for (int ci = 0; ci < 2; ++ci)
#pragma unroll
      for (int dh = 0; dh < 3; ++dh)
#pragma unroll
        for (int dw = 0; dw < 3; ++dw) {
          int hh = h + dh - 1, ww = w + dw - 1;
          bool ok = (hh >= 0) & (hh < 64) & (ww >= 0) & (ww < 64);
          xin[ci][dh][dw] = ok ? xn[ci * 4096 + hh * 64 + ww] : 0.0f;
        }
#pragma unroll
    for (int c = 0; c < 16; ++c) {
      float acc = 0.f;
#pragma unroll
      for (int ci = 0; ci < 2; ++ci)
#pragma unroll
        for (int kh = 0; kh < 3; ++kh)
#pragma unroll
          for (int kw = 0; kw < 3; ++kw)
            acc = fmaf(xin[ci][kh][kw], sw[(c * 2 + ci) * 9 + kh * 3 + kw], acc);
      sum[c] += acc;
      sq[c]  += acc * acc;
    }
  }
#pragma unroll
  for (int c = 0; c < 16; ++c) {
    buf[c * 257 + tid]        = sum[c];
    buf[(16 + c) * 257 + tid] = sq[c];
  }
  __syncthreads();
  if (tid < 32) {                       // fixed-order sum -> deterministic
    float a = 0.f;
    for (int i = 0; i < 256; ++i) a += buf[tid * 257 + i];
    partial[n * 32 + tid] = a;
  }
}

__global__ void reduce_stats_kernel(const float* __restrict__ partial,
                                    float* __restrict__ stats) {
  int c = threadIdx.x;
  if (c < 32) {
    float a = 0.f;
    for (int n = 0; n < B_; ++n) a += partial[n * 32 + c];
    stats[c] = a;
  }
}

__global__ __launch_bounds__(256) void encoder_kernel(const float* __restrict__ x,
    const float* __restrict__ cw, const float* __restrict__ stats,
    const float* __restrict__ gamma, const float* __restrict__ beta,
    const float* __restrict__ fcw, const float* __restrict__ fcb,
    float* __restrict__ out, float* __restrict__ xbuf) {
  __shared__ float sw[288];
  __shared__ float scale_s[16], shift_s[16];
  __shared__ float wred[8][16];
  __shared__ float feat[16];
  const int tid = threadIdx.x;
  const int n   = blockIdx.x;
  for (int i = tid; i < 288; i += 256) sw[i] = cw[i];
  if (tid < 16) {
    const float inv = 1.0f / (512.0f * 4096.0f);
    float mean = stats[tid] * inv;
    float var  = stats[16 + tid] * inv - mean * mean;
    float sc   = gamma[tid] * rsqrtf(var + 1e-5f);
    scale_s[tid] = sc;
    shift_s[tid] = beta[tid] - mean * sc;
  }
  __syncthreads();
  const float* xn = x + (size_t)n * 2 * 4096;
  float fsum[16];
#pragma unroll
  for (int c = 0; c < 16; ++c) fsum[c] = 0.f;
  for (int pp = tid; pp < 1024; pp += 256) {
    int ph = pp >> 5, pw = pp & 31;
    int h0 = ph * 2 - 1, w0 = pw * 2 - 1;
    float xin[2][4][4];
#pragma unroll
    for (int ci = 0; ci < 2; ++ci)
#pragma unroll
      for (int r = 0; r < 4; ++r)
#pragma unroll
        for (int cc = 0; cc < 4; ++cc) {
          int hh = h0 + r, ww = w0 + cc;
          bool ok = (hh >= 0) & (hh < 64) & (ww >= 0) & (ww < 64);
          xin[ci][r][cc] = ok ? xn[ci * 4096 + hh * 64 + ww] : 0.0f;
        }
#pragma unroll
    for (int c = 0; c < 16; ++c) {
      float mx = 0.0f;
#pragma unroll
      for (int dy = 0; dy < 2; ++dy)
#pragma unroll
        for (int dx = 0; dx < 2; ++dx) {
          float acc = 0.f;
#pragma unroll
          for (int ci = 0; ci < 2; ++ci)
#pragma unroll
            for (int kh = 0; kh < 3; ++kh)
#pragma unroll
              for (int kw = 0; kw < 3; ++kw)
                acc = fmaf(xin[ci][dy + kh][dx + kw], sw[(c * 2 + ci) * 9 + kh * 3 + kw], acc);
          float v = fmaf(acc, scale_s[c], shift_s[c]);
          v  = fmaxf(v, 0.0f);
          mx = fmaxf(mx, v);
        }
      fsum[c] += mx;
    }
  }
  const int lane = tid & 31, wave = tid >> 5;
#pragma unroll
  for (int c = 0; c < 16; ++c) {
    float v = fsum[c];
#pragma unroll
    for (int off = 16; off > 0; off >>= 1) v += __shfl_xor(v, off, 32);
    if (lane == 0) wred[wave][c] = v;
  }
  __syncthreads();
  if (tid < 16) {
    float a = 0.f;
#pragma unroll
    for (int w = 0; w < 8; ++w) a += wred[w][tid];
    feat[tid] = a * (1.0f / 1024.0f);
  }
  __syncthreads();
  {
    int r = tid;
    float acc = fcb[r];
#pragma unroll
    for (int c = 0; c < 16; ++c) acc = fmaf(feat[c], fcw[r * 16 + c], acc);
    out[(size_t)n * OUT_STRIDE + r] = acc;  // t = 0 slot
    xbuf[n * XSTRIDE + r] = acc;            // s_prev for step 0
  }
  if (tid < 4) xbuf[n * XSTRIDE + 256 + tid] = 0.0f;
}

// ---------------------------------------------------------------- GRU pieces
__global__ void set_u_kernel(const float* __restrict__ actions,
                             float* __restrict__ xbuf, int t) {
  int n = blockIdx.x * blockDim.x + threadIdx.x;
  if (n < B_) {
    xbuf[n * XSTRIDE + 256] = actions[n * (TSTEPS * 2) + t * 2 + 0];
    xbuf[n * XSTRIDE + 257] = actions[n * (TSTEPS * 2) + t * 2 + 1];
  }
}

// C[M,N] = A[M,K] @ W[N,K]^T + bias  via V_WMMA_F32_16X16X4_F32.
// Block = 4 waves, covers (16*MT) rows x 256 cols. The A tile (shared by all
// waves) is DMA'd into LDS by the Tensor Data Mover with tile_dim0 = Kpad >
// tensor_dim0 = K so TDM's OOB zero-fill creates a conflict-free LDS pitch.
// k-loop is software pipelined (load k+4 fragments before k's WMMAs).
template <int MT>
__global__ __launch_bounds__(128) void gemm_wmma_kernel(
    const float* __restrict__ A, int lda,
    const float* __restrict__ W, int ldw,
    const float* __restrict__ bias,
    float* __restrict__ C,  int ldc,
    float* __restrict__ C2, int ldc2,
    int K, int Kpad, int doRelu) {
  extern __shared__ float As[];                 // (16*MT) * Kpad floats
  const int lane = threadIdx.x & 31;
  const int wave = threadIdx.x >> 5;
  const int lo = lane & 15, hi = lane >> 4;
  const int m0 = blockIdx.x * (16 * MT);
  const int n0 = blockIdx.y * 256 + wave * 64;

  // ---- TDM: stage A tile (16*MT rows x K cols, pitch Kpad) into LDS ------
  if (threadIdx.x < 32) {                       // wave 0 issues the DMA
    unsigned lds_off = (unsigned)(size_t)(&As[0]);
    unsigned long long ga = (unsigned long long)(size_t)(A + (size_t)m0 * lda);
    u32x4 g0;
    g0[0] = 1u;                                            // count=1, user desc
    g0[1] = lds_off;                                       // lds_addr
    g0[2] = (unsigned)(ga & 0xFFFFFFFFull);                // global_addr[31:0]
    g0[3] = (unsigned)((ga >> 32) & 0x1FFFFFFull) | (2u << 30); // [56:32] | type=2
    const unsigned td0 = (unsigned)K;           // tensor_dim0: valid cols
    const unsigned td1 = (unsigned)(16 * MT);   // tensor_dim1: rows
    const unsigned tl0 = (unsigned)Kpad;        // tile_dim0 (> td0 -> zero fill)
    const unsigned tl1 = (unsigned)(16 * MT);   // tile_dim1
    i32x8 g1;
    g1[0] = (int)(2u << 16);                    // data_size = 4 bytes
    g1[1] = (int)((td0 & 0xFFFFu) << 16);       // dim0[15:0] at bits 63:48
    g1[2] = (int)((td0 >> 16) | ((td1 & 0xFFFFu) << 16));
    g1[3] = (int)((td1 >> 16) | (tl0 << 16));   // tile_dim0 at bits 127:112
    g1[4] = (int)tl1;                           // tile_dim1 (tile_dim2 = 0)
    g1[5] = (int)(unsigned)lda;                 // tensor_dim0_stride[31:0]
    g1[6] = 0;
    g1[7] = 0;
    i32x4 z4 = {0, 0, 0, 0};
    i32x8 z8 = {0, 0, 0, 0, 0, 0, 0, 0};
    __builtin_amdgcn_tensor_load_to_lds(g0, g1, z4, z4, z8, 0);
    __builtin_amdgcn_s_wait_tensorcnt(0);
  }
  __syncthreads();

  // ---- accumulators + weight pointers ------------------------------------
  v8f acc[MT][4];
#pragma unroll
  for (int i = 0; i < MT; ++i)
#pragma unroll
    for (int j = 0; j < 4; ++j) acc[i][j] = (v8f){0, 0, 0, 0, 0, 0, 0, 0};

  const float* Wp[4];
#pragma unroll
  for (int j = 0; j < 4; ++j)
    Wp[j] = W + (size_t)(n0 + 16 * j + lo) * ldw + 2 * hi;

  const int kk0 = 2 * hi;
  // prologue: fragments for k = 0
  v2f aC[MT], bC[4];
#pragma unroll
  for (int i = 0; i < MT; ++i) {
    aC[i].x = As[(i * 16 + lo) * Kpad + kk0];
    aC[i].y = As[(i * 16 + lo) * Kpad + kk0 + 1];
  }
#pragma unroll
  for (int j = 0; j < 4; ++j) { bC[j].x = Wp[j][0]; bC[j].y = Wp[j][1]; }

#pragma unroll 2
  for (int k = 4; k < K; k += 4) {
    v2f aN[MT], bN[4];
#pragma unroll
    for (int j = 0; j < 4; ++j) { bN[j].x = Wp[j][k]; bN[j].y = Wp[j][k + 1]; }
#pragma unroll
    for (int i = 0; i < MT; ++i) {
      aN[i].x = As[(i * 16 + lo) * Kpad + k + kk0];
      aN[i].y = As[(i * 16 + lo) * Kpad + k + kk0 + 1];
    }
#pragma unroll
    for (int i = 0; i < MT; ++i)
#pragma unroll
      for (int j = 0; j < 4; ++j) acc[i][j] = wmma4(aC[i], bC[j], acc[i][j]);
#pragma unroll
    for (int i = 0; i < MT; ++i) aC[i] = aN[i];
#pragma unroll
    for (int j = 0; j < 4; ++j) bC[j] = bN[j];
  }
#pragma unroll
  for (int i = 0; i < MT; ++i)
#pragma unroll
    for (int j = 0; j < 4; ++j) acc[i][j] = wmma4(aC[i], bC[j], acc[i][j]);

  // ---- epilogue: bias (+relu) and store ----------------------------------
#pragma unroll
  for (int i = 0; i < MT; ++i)
#pragma unroll
    for (int j = 0; j < 4; ++j) {
      int nc = n0 + 16 * j + lo;
      float bv = bias[nc];
#pragma unroll
      for (int r = 0; r < 8; ++r) {
        int mr = m0 + 16 * i + r + 8 * hi;
        float v = acc[i][j][r] + bv;
        if (doRelu) v = fmaxf(v, 0.0f);
        C[(size_t)mr * ldc + nc] = v;
        if (C2) C2[(size_t)mr * ldc2 + nc] = v;
      }
    }
}

__global__ __launch_bounds__(256) void gru_pointwise_kernel(const float* __restrict__ gx,
    const float* __restrict__ gh, float* __restrict__ h) {
  int idx = blockIdx.x * 256 + threadIdx.x;       // 512*512 elements
  int m = idx >> 9, j = idx & 511;
  const float* gxm = gx + m * 1536;
  const float* ghm = gh + m * 1536;
  float r  = 1.0f / (1.0f + expf(-(gxm[j] + ghm[j])));
  float z  = 1.0f / (1.0f + expf(-(gxm[512 + j] + ghm[512 + j])));
  float nn = tanhf(gxm[1024 + j] + r * ghm[1024 + j]);
  float hp = h[idx];
  h[idx] = (1.0f - z) * nn + z * hp;
}

// ---------------------------------------------------------------- launcher
extern "C" void kernel_launch(void* const* d_in, const int* in_sizes, int n_in,
                              void* d_out, int out_size, void* d_ws, size_t ws_size,
                              hipStream_t stream) {
  (void)in_sizes; (void)n_in; (void)out_size; (void)ws_size;
  const float* init_state = (const float*)d_in[0];
  const float* actions    = (const float*)d_in[1];
  const float* conv_w     = (const float*)d_in[2];
  const float* bn_gamma   = (const float*)d_in[3];
  const float* bn_beta    = (const float*)d_in[4];
  const float* enc_fc_w   = (const float*)d_in[5];
  const float* enc_fc_b   = (const float*)d_in[6];
  const float* gru_w_ih   = (const float*)d_in[7];
  const float* gru_w_hh   = (const float*)d_in[8];
  const float* gru_b_ih   = (const float*)d_in[9];
  const float* gru_b_hh   = (const float*)d_in[10];
  const float* pred_fc_w  = (const float*)d_in[11];
  const float* pred_fc_b  = (const float*)d_in[12];
  float* out = (float*)d_out;

  float* ws      = (float*)d_ws;
  float* xbuf    = ws;                        // 512*260
  float* h       = xbuf + B_ * XSTRIDE;       // 512*512
  float* gx      = h    + B_ * HID_;          // 512*1536
  float* gh      = gx   + B_ * 3 * HID_;      // 512*1536
  float* wihp    = gh   + B_ * 3 * HID_;      // 1536*260
  float* stats   = wihp + 3 * HID_ * XSTRIDE; // 32
  float* partial = stats + 32;                // 512*32

  // Allow >64KB dynamic LDS for the MT=2 instantiation (gh: 32*514*4 B).
  (void)hipFuncSetAttribute(reinterpret_cast<const void*>(&gemm_wmma_kernel<2>),
                            hipFuncAttributeMaxDynamicSharedMemorySize, 131072);
  (void)hipFuncSetAttribute(reinterpret_cast<const void*>(&gemm_wmma_kernel<1>),
                            hipFuncAttributeMaxDynamicSharedMemorySize, 131072);

  zero_kernel<<<(B_ * HID_ + 255) / 256, 256, 0, stream>>>(h, B_ * HID_);
  pack_wih_kernel<<<(3 * HID_ * XSTRIDE + 255) / 256, 256, 0, stream>>>(gru_w_ih, wihp);
  conv_stats_kernel<<<B_, 256, 0, stream>>>(init_state, conv_w, partial);
  reduce_stats_kernel<<<1, 32, 0, stream>>>(partial, stats);
  encoder_kernel<<<B_, 256, 0, stream>>>(init_state, conv_w, stats, bn_gamma, bn_beta,
                                         enc_fc_w, enc_fc_b, out, xbuf);

  const int KPX = XSTRIDE + 2;   // 262: LDS pitch for K=260
  const int KPH = HID_ + 2;      // 514: LDS pitch for K=512
  dim3 blk(128);
  dim3 grid_g(B_ / 32, (3 * HID_) / 256);   // 16 x 6  (MT=2)
  dim3 grid_p(B_ / 16, REPR_ / 256);        // 32 x 1  (MT=1)
  const size_t sh_gx   = (size_t)32 * KPX * sizeof(float);  // 33,536 B
  const size_t sh_gh   = (size_t)32 * KPH * sizeof(float);  // 65,792 B
  const size_t sh_pred = (size_t)16 * KPH * sizeof(float);  // 32,896 B

  for (int t = 0; t < TSTEPS; ++t) {
    set_u_kernel<<<2, 256, 0, stream>>>(actions, xbuf, t);
    gemm_wmma_kernel<2><<<grid_g, blk, sh_gx, stream>>>(
        xbuf, XSTRIDE, wihp, XSTRIDE, gru_b_ih, gx, 3 * HID_, nullptr, 0,
        XSTRIDE, KPX, 0);
    gemm_wmma_kernel<2><<<grid_g, blk, sh_gh, stream>>>(
        h, HID_, gru_w_hh, HID_, gru_b_hh, gh, 3 * HID_, nullptr, 0,
        HID_, KPH, 0);
    gru_pointwise_kernel<<<(B_ * HID_) / 256, 256, 0, stream>>>(gx, gh, h);
    gemm_wmma_kernel<1><<<grid_p, blk, sh_pred, stream>>>(
        h, HID_, pred_fc_w, HID_, pred_fc_b,
        out + (size_t)(t + 1) * REPR_, OUT_STRIDE, xbuf, XSTRIDE,
        HID_, KPH, 1);
  }
}